// Downsample_87325275062838
// MI455X (gfx1250) — compile-verified
//
#include <hip/hip_runtime.h>

// Downsample (upfirdn2d: 4x4 FIR, stride-2, pad 1) for x:(8,128,256,256) f32.
// Pure HBM-bandwidth problem (~320MB traffic -> ~14us floor @23.3TB/s).
// Strategy: stage full-width row tiles into LDS with CDNA5 async B128
// global->LDS loads (ASYNCcnt path), compute branch-free from zero-padded LDS,
// emit coalesced float4 stores.

#define H        256
#define W        256
#define OH       128
#define OW       128
#define TILE_OR  16                      // output rows per block
#define IN_ROWS  (2 * TILE_OR + 2)       // 34 input rows (incl. vertical halo)
#define LDS_STRIDE 264                   // floats per LDS row (16B-aligned data @ +4)
#define NPLANES  (8 * 128)
#define THREADS  512

#if defined(__gfx1250__) && __has_builtin(__builtin_amdgcn_global_load_async_to_lds_b128)
#define USE_ASYNC_LDS 1
#else
#define USE_ASYNC_LDS 0
#endif

#if USE_ASYNC_LDS
// Builtin signature (from hipcc diagnostic): param 0 is
// 'int __attribute__((vector_size(16))) __device__ *' i.e. AS(1) v4i*.
typedef int v4i __attribute__((vector_size(16)));
typedef __attribute__((address_space(1))) v4i GlobalV4i;
typedef __attribute__((address_space(3))) v4i LdsV4i;
#endif

__global__ __launch_bounds__(THREADS) void Downsample_87325275062838_kernel(
    const float* __restrict__ x, const float* __restrict__ kern,
    float* __restrict__ out)
{
    __shared__ float lds[IN_ROWS * LDS_STRIDE];  // 34*264*4 = 35,904 B

    const int tid      = threadIdx.x;
    const int plane    = blockIdx.y;                 // 0..1023 (b*c)
    const int row0_out = blockIdx.x * TILE_OR;       // first output row of tile
    const int row0_in  = 2 * row0_out - 1;           // first input row (may be -1)
    const float* __restrict__ src = x + (size_t)plane * (H * W);

    // Zero the horizontal halo columns (input col -1 -> off 3, col 256 -> off 260).
    if (tid < 2 * IN_ROWS) {
        const int r    = tid >> 1;
        const int side = tid & 1;
        lds[r * LDS_STRIDE + (side ? 260 : 3)] = 0.0f;
    }

    // Stage IN_ROWS x 256 floats as 64 float4 chunks/row. Global col c maps to
    // LDS offset r*LDS_STRIDE + 4 + c (16B aligned per chunk).
    for (int i = tid; i < IN_ROWS * 64; i += THREADS) {
        const int r     = i >> 6;        // 0..33
        const int chunk = i & 63;        // 0..63
        const int grow  = row0_in + r;   // global input row, may be OOB
        const int loff  = r * LDS_STRIDE + 4 + chunk * 4;
        if ((unsigned)grow < (unsigned)H) {
            const float* g = src + (size_t)grow * W + chunk * 4;
#if USE_ASYNC_LDS
            __builtin_amdgcn_global_load_async_to_lds_b128(
                (GlobalV4i*)g, (LdsV4i*)&lds[loff], /*offset=*/0, /*cpol=*/0);
#else
            *reinterpret_cast<float4*>(&lds[loff]) =
                *reinterpret_cast<const float4*>(g);
#endif
        } else {
            float4 z = make_float4(0.f, 0.f, 0.f, 0.f);
            *reinterpret_cast<float4*>(&lds[loff]) = z;
        }
    }

#if USE_ASYNC_LDS
#if __has_builtin(__builtin_amdgcn_s_wait_asynccnt)
    __builtin_amdgcn_s_wait_asynccnt(0);
#else
    asm volatile("s_wait_asynccnt 0" ::: "memory");
#endif
#endif
    __syncthreads();

    // JAX reference flips the kernel then correlates: effective tap (ky,kx)
    // weight is kernel[3-ky][3-kx] == kern[15 - (ky*4+kx)].
    float wf[16];
#pragma unroll
    for (int i = 0; i < 16; ++i) wf[i] = kern[15 - i];

    // 512 threads: orow = tid>>5 (0..15), 4 consecutive output cols per thread.
    const int orow = tid >> 5;
    const int oc0  = (tid & 31) << 2;    // 0..124

    float acc0 = 0.f, acc1 = 0.f, acc2 = 0.f, acc3 = 0.f;
#pragma unroll
    for (int ky = 0; ky < 4; ++ky) {
        // input col (2*oc0 - 1 + t) lives at offset base + t
        const float* rp = &lds[(2 * orow + ky) * LDS_STRIDE + 3 + 2 * oc0];
        float v[10];
#pragma unroll
        for (int t = 0; t < 10; ++t) v[t] = rp[t];
        const float w0 = wf[ky * 4 + 0], w1 = wf[ky * 4 + 1];
        const float w2 = wf[ky * 4 + 2], w3 = wf[ky * 4 + 3];
        acc0 += w0 * v[0] + w1 * v[1] + w2 * v[2] + w3 * v[3];
        acc1 += w0 * v[2] + w1 * v[3] + w2 * v[4] + w3 * v[5];
        acc2 += w0 * v[4] + w1 * v[5] + w2 * v[6] + w3 * v[7];
        acc3 += w0 * v[6] + w1 * v[7] + w2 * v[8] + w3 * v[9];
    }

    float4 o = make_float4(acc0, acc1, acc2, acc3);
    *reinterpret_cast<float4*>(
        out + (size_t)plane * (OH * OW) + (size_t)(row0_out + orow) * OW + oc0) = o;
}

extern "C" void kernel_launch(void* const* d_in, const int* in_sizes, int n_in,
                              void* d_out, int out_size, void* d_ws, size_t ws_size,
                              hipStream_t stream) {
    (void)in_sizes; (void)n_in; (void)out_size; (void)d_ws; (void)ws_size;
    const float* x    = (const float*)d_in[0];   // 8*128*256*256 f32
    const float* kern = (const float*)d_in[1];   // 4*4 f32 (normalized blur)
    float* out        = (float*)d_out;           // 8*128*128*128 f32

    dim3 grid(OH / TILE_OR, NPLANES);            // (8, 1024)
    dim3 block(THREADS);                         // 512 = 16 wave32
    Downsample_87325275062838_kernel<<<grid, block, 0, stream>>>(x, kern, out);
}